// MessagePassing_70901320122389
// MI455X (gfx1250) — compile-verified
//
#include <hip/hip_runtime.h>
#include <math.h>

// ---------------------------------------------------------------------------
// Equivariant 2-layer message passing (NequIP-style) for MI455X / gfx1250.
// Edge MLP (the dominant dense GEMM) runs on V_WMMA_F32_16X16X4_F32 and is
// fused with the gather / tensor-product / atomic-scatter epilogue so the
// E x 192 edge-weight tensor never touches HBM.
// ---------------------------------------------------------------------------

typedef float v2f __attribute__((ext_vector_type(2)));
typedef float v4f __attribute__((ext_vector_type(4)));
typedef float v8f __attribute__((ext_vector_type(8)));

#define CS 64
#define CV 32
#define CM 96          // CS + CV
#define WN 192         // 2*CS + 2*CV
#define FCH 64

#define INV8        0.125f
#define INV_SQRT32  0.1767766952966369f
#define INV_SQRT96  0.1020620726159658f
#define INV_SQRT3   0.5773502691896258f

__device__ __forceinline__ float sigmoidf_(float x) { return 1.0f / (1.0f + __expf(-x)); }
__device__ __forceinline__ float siluf_(float x)    { return x / (1.0f + __expf(-x)); }

// ---------------------------------------------------------------------------
// One 16x16 C tile of A(LDS, row-major) x B(global, row-major), K multiple of 4,
// using V_WMMA_F32_16X16X4_F32.
// A fragment layout (ISA 7.12.2, 32-bit A 16x4): lane l<16 holds A[l][k],A[l][k+1];
// lane l+16 holds A[l][k+2],A[l][k+3].  B mirrors with N across lanes.
// C layout: vgpr i -> row i (lanes 0-15) / row 8+i (lanes 16-31), col = lane&15.
// ---------------------------------------------------------------------------
__device__ __forceinline__ v8f wmma_tile_f32(const float* __restrict__ Alds, int lda, int m0,
                                             const float* __restrict__ Bg,   int ldb, int n0,
                                             int K, int lane)
{
    v8f acc = {};
    const int l  = lane & 15;
    const int kh = (lane >> 4) << 1;   // 0 for lanes 0-15, 2 for lanes 16-31
    for (int k = 0; k < K; k += 4) {
        v2f a, b;
        a[0] = Alds[(m0 + l) * lda + k + kh];
        a[1] = Alds[(m0 + l) * lda + k + kh + 1];
        b[0] = Bg[(size_t)(k + kh) * ldb + n0 + l];
        b[1] = Bg[(size_t)(k + kh + 1) * ldb + n0 + l];
        acc = __builtin_amdgcn_wmma_f32_16x16x4_f32(false, a, false, b, (short)0, acc, false, false);
    }
    return acc;
}

// ---------------------------------------------------------------------------
// degree + 1/sqrt(deg)
// ---------------------------------------------------------------------------
__global__ void deg_kernel(const int* __restrict__ edge_dst, float* __restrict__ deg, int E)
{
    int e = blockIdx.x * 256 + threadIdx.x;
    if (e < E) atomicAdd(&deg[edge_dst[e]], 1.0f);
}

__global__ void isd_kernel(const float* __restrict__ deg, float* __restrict__ isd, int N)
{
    int n = blockIdx.x * 256 + threadIdx.x;
    if (n < N) isd[n] = rsqrtf(fmaxf(deg[n], 1.0f));
}

// ---------------------------------------------------------------------------
// node_front: sc_s = s@Wsc_s*attr/8 ; f_s = s@Wl1_s*attr/8
//             sc_v[n][w][c] = sum_u v[n][u][c]*Wsc_v[u][w]*attr/sqrt(32); f_v likewise
// 16 nodes / 256-thread block; weights staged in LDS.
// ---------------------------------------------------------------------------
__global__ __launch_bounds__(256)
void node_front_kernel(const float* __restrict__ sp, int ss,
                       const float* __restrict__ vp, int vs,
                       const float* __restrict__ attr,
                       const float* __restrict__ Wscs, int scw,
                       const float* __restrict__ Wscv,
                       const float* __restrict__ Wl1s,
                       const float* __restrict__ Wl1v,
                       float* __restrict__ sc_s,   // N x 96 (stride 96, scw valid)
                       float* __restrict__ sc_v,   // N x 96
                       float* __restrict__ f_s,    // N x 64
                       float* __restrict__ f_v,    // N x 96
                       int N)
{
    __shared__ float sWscs[CS * CM];       // up to 64x96
    __shared__ float sWl1s[CS * CS];
    __shared__ float sWscv[CV * CV];
    __shared__ float sWl1v[CV * CV];
    __shared__ float sS[16 * CS];
    __shared__ float sV[16 * CM];

    const int tid = threadIdx.x;
    const int n0  = blockIdx.x * 16;

    for (int i = tid; i < CS * scw; i += 256) sWscs[i] = Wscs[i];
    for (int i = tid; i < CS * CS;  i += 256) sWl1s[i] = Wl1s[i];
    for (int i = tid; i < CV * CV;  i += 256) { sWscv[i] = Wscv[i]; sWl1v[i] = Wl1v[i]; }
    for (int i = tid; i < 16 * CS; i += 256) {
        int r = i >> 6, c = i & 63, n = n0 + r;
        sS[i] = (n < N) ? sp[(size_t)n * ss + c] : 0.0f;
    }
    for (int i = tid; i < 16 * CM; i += 256) {
        int r = i / CM, k = i % CM, n = n0 + r;
        sV[i] = (n < N) ? vp[(size_t)n * vs + k] : 0.0f;
    }
    __syncthreads();

    // scalar-channel outputs
    const int stot = scw + CS;
    for (int idx = tid; idx < 16 * stot; idx += 256) {
        int r = idx / stot, j = idx % stot, n = n0 + r;
        if (n >= N) continue;
        float a = attr[n];
        if (j < scw) {
            float acc = 0.0f;
            for (int k = 0; k < CS; ++k) acc += sS[r * CS + k] * sWscs[k * scw + j];
            sc_s[(size_t)n * CM + j] = acc * a * INV8;
        } else {
            int j2 = j - scw;
            float acc = 0.0f;
            for (int k = 0; k < CS; ++k) acc += sS[r * CS + k] * sWl1s[k * CS + j2];
            f_s[(size_t)n * CS + j2] = acc * a * INV8;
        }
    }
    // vector-channel outputs
    for (int idx = tid; idx < 16 * 2 * CM; idx += 256) {
        int r = idx / (2 * CM), j = idx % (2 * CM), n = n0 + r;
        if (n >= N) continue;
        float a = attr[n] * INV_SQRT32;
        int jj = (j < CM) ? j : (j - CM);
        int w = jj / 3, c = jj % 3;
        const float* Wv = (j < CM) ? sWscv : sWl1v;
        float acc = 0.0f;
        for (int u = 0; u < CV; ++u) acc += sV[r * CM + u * 3 + c] * Wv[u * CV + w];
        if (j < CM) sc_v[(size_t)n * CM + jj] = acc * a;
        else        f_v[(size_t)n * CM + jj]  = acc * a;
    }
}

// ---------------------------------------------------------------------------
// edge kernel: per 256-thread block, a 64-edge tile.
//   X(64x64)=edge_scalars tile -> LDS
//   H = silu(X@Wfa/8)   (WMMA f32, 4x4 tiles, 8 waves)
//   W = H@Wfb/8         (WMMA f32, 4x12 tiles) -> LDS (cols 0-63 in sX, 64-191 in sW)
//   epilogue: gather f[src], build mid_s/mid_v, atomic-scatter into agg[dst]
// ---------------------------------------------------------------------------
__global__ __launch_bounds__(256)
void edge_kernel(const float* __restrict__ edge_scalars,
                 const float* __restrict__ edge_attr,
                 const int*   __restrict__ edge_src,
                 const int*   __restrict__ edge_dst,
                 const float* __restrict__ f_s,
                 const float* __restrict__ f_v,
                 const float* __restrict__ Wfa,   // 64x64
                 const float* __restrict__ Wfb,   // 64x192
                 float* __restrict__ agg_s,       // N x 96
                 float* __restrict__ agg_v,       // N x 288
                 int E)
{
    __shared__ float sX[64 * 64];    // X tile; later w columns 0..63
    __shared__ float sH[64 * 64];    // hidden tile
    __shared__ float sW[64 * 128];   // w columns 64..191

    const int tid  = threadIdx.x;
    const int wave = tid >> 5;
    const int lane = tid & 31;
    const int e0   = blockIdx.x * 64;

    // load X tile (zero-pad past E)
    for (int i = 0; i < 4; ++i) {
        int q    = tid + i * 256;           // float4 index 0..1023
        int row  = q >> 4;
        int col4 = (q & 15) << 2;
        int e    = e0 + row;
        v4f val = {};
        if (e < E) val = *(const v4f*)(edge_scalars + (size_t)e * CS + col4);
        *(v4f*)(sX + row * 64 + col4) = val;
    }
    __syncthreads();

    // GEMM1: H = silu(X @ Wfa / 8); 16 tiles, 2 per wave (uniform)
    for (int t = wave; t < 16; t += 8) {
        int m0 = (t >> 2) * 16, n0 = (t & 3) * 16;
        v8f acc = wmma_tile_f32(sX, 64, m0, Wfa, 64, n0, 64, lane);
        int col   = n0 + (lane & 15);
        int rbase = m0 + ((lane >> 4) << 3);
        for (int i = 0; i < 8; ++i) {
            float c = acc[i] * INV8;
            sH[(rbase + i) * 64 + col] = siluf_(c);
        }
    }
    __syncthreads();

    // GEMM2: W = H @ Wfb / 8; 48 tiles, 6 per wave (uniform)
    for (int t = wave; t < 48; t += 8) {
        int m0 = (t & 3) * 16, n0 = (t >> 2) * 16;
        v8f acc = wmma_tile_f32(sH, 64, m0, Wfb, WN, n0, 64, lane);
        int col   = n0 + (lane & 15);
        int rbase = m0 + ((lane >> 4) << 3);
        for (int i = 0; i < 8; ++i) {
            float wv = acc[i] * INV8;
            int r = rbase + i;
            if (col < 64) sX[r * 64 + col]        = wv;
            else          sW[r * 128 + (col - 64)] = wv;
        }
    }
    __syncthreads();

    // epilogue: 4 threads per edge
    const int el  = tid >> 2;
    const int sub = tid & 3;
    const int e   = e0 + el;
    if (e < E) {
        const int src = edge_src[e];
        const int dst = edge_dst[e];
        const float eas = edge_attr[(size_t)e * 4 + 0];
        const float ev0 = edge_attr[(size_t)e * 4 + 1];
        const float ev1 = edge_attr[(size_t)e * 4 + 2];
        const float ev2 = edge_attr[(size_t)e * 4 + 3];
        const float* w0 = sX + el * 64;     // w cols 0..63   (w00)
        const float* w1 = sW + el * 128;    // w cols 64..191 (w01|w10|w11)
        float* as = agg_s + (size_t)dst * CM;
        float* av = agg_v + (size_t)dst * (CM * 3);
        const float* gsrow = f_s + (size_t)src * CS;
        const float* gvrow = f_v + (size_t)src * CM;

        // scalar channels k in [sub*16, sub*16+16)
        for (int k = sub * 16; k < sub * 16 + 16; ++k) {
            float gs  = gsrow[k];
            float m00 = w0[k] * gs * eas;           // mid_s[:CS]
            atomicAdd(&as[k], m00);
            float t01 = w1[k] * gs;                  // w01 = col 64+k
            atomicAdd(&av[k * 3 + 0], t01 * ev0);    // mid_v[:CS]
            atomicAdd(&av[k * 3 + 1], t01 * ev1);
            atomicAdd(&av[k * 3 + 2], t01 * ev2);
        }
        // vector channels u in [sub*8, sub*8+8)
        for (int u = sub * 8; u < sub * 8 + 8; ++u) {
            float g0 = gvrow[u * 3 + 0];
            float g1 = gvrow[u * 3 + 1];
            float g2 = gvrow[u * 3 + 2];
            float w10 = w1[64 + u];                  // col 128+u
            float w11 = w1[96 + u];                  // col 160+u
            float dv = g0 * ev0 + g1 * ev1 + g2 * ev2;
            atomicAdd(&as[CS + u], w11 * dv * INV_SQRT3);
            float t10 = w10 * eas;
            atomicAdd(&av[(CS + u) * 3 + 0], t10 * g0);
            atomicAdd(&av[(CS + u) * 3 + 1], t10 * g1);
            atomicAdd(&av[(CS + u) * 3 + 2], t10 * g2);
        }
    }
}

// ---------------------------------------------------------------------------
// node_back: agg *= isd; out_s = agg_s@W2s*attr/sqrt96; out_v likewise;
// angle = 0.1*agg_s@W3*attr/sqrt96; h = cos*sc + sin*out.
// layer 0 -> g_s = silu(h_s[:64]), g_v = sigmoid(h_s[64:96])*h_v
// layer 1 -> d_out[n] = [o_s(64) | o_v(96)]
// 8 nodes / 256-thread block.
// ---------------------------------------------------------------------------
__global__ __launch_bounds__(256)
void node_back_kernel(const float* __restrict__ agg_s,
                      const float* __restrict__ agg_v,
                      const float* __restrict__ isd,
                      const float* __restrict__ attr,
                      const float* __restrict__ sc_s,   // N x 96 (scw valid)
                      const float* __restrict__ sc_v,   // N x 96
                      const float* __restrict__ W2s, int scw,  // 96 x scw
                      const float* __restrict__ W2v,           // 96 x 32
                      const float* __restrict__ W3,            // 96
                      float* __restrict__ out_a,   // layer0: g_s (Nx64); layer1: d_out (Nx160)
                      float* __restrict__ out_b,   // layer0: g_v (Nx96); layer1: unused
                      int N, int final_layer)
{
    __shared__ float sW2s[CM * CM];
    __shared__ float sW2v[CM * CV];
    __shared__ float sW3[CM];
    __shared__ float sAs[8 * CM];
    __shared__ float sAv[8 * CM * 3];
    __shared__ float sHv[8 * CM];
    __shared__ float sCos[8];
    __shared__ float sSin[8];

    const int tid = threadIdx.x;
    const int n0  = blockIdx.x * 8;

    for (int i = tid; i < CM * scw; i += 256) sW2s[i] = W2s[i];
    for (int i = tid; i < CM * CV;  i += 256) sW2v[i] = W2v[i];
    for (int i = tid; i < CM;       i += 256) sW3[i]  = W3[i];
    for (int i = tid; i < 8 * CM; i += 256) {
        int r = i / CM, k = i % CM, n = n0 + r;
        sAs[i] = (n < N) ? agg_s[(size_t)n * CM + k] * isd[n] : 0.0f;
    }
    for (int i = tid; i < 8 * CM * 3; i += 256) {
        int r = i / (CM * 3), k = i % (CM * 3), n = n0 + r;
        sAv[i] = (n < N) ? agg_v[(size_t)n * CM * 3 + k] * isd[n] : 0.0f;
    }
    __syncthreads();

    if (tid < 8) {
        int n = n0 + tid;
        float cA = 1.0f, sA = 0.0f;
        if (n < N) {
            float d = 0.0f;
            for (int m = 0; m < CM; ++m) d += sAs[tid * CM + m] * sW3[m];
            float ang = 0.1f * d * attr[n] * INV_SQRT96;
            cA = cosf(ang); sA = sinf(ang);
        }
        sCos[tid] = cA; sSin[tid] = sA;
    }
    __syncthreads();

    // vector part: h_v into LDS
    for (int idx = tid; idx < 8 * CM; idx += 256) {
        int r = idx / CM, j = idx % CM, n = n0 + r;
        if (n >= N) continue;
        int w = j / 3, c = j % 3;
        float acc = 0.0f;
        for (int m = 0; m < CM; ++m) acc += sAv[r * CM * 3 + m * 3 + c] * sW2v[m * CV + w];
        float outv = acc * attr[n] * INV_SQRT96;
        sHv[r * CM + j] = sCos[r] * sc_v[(size_t)n * CM + j] + sSin[r] * outv;
    }
    __syncthreads();

    // scalar part + gating / output
    for (int idx = tid; idx < 8 * scw; idx += 256) {
        int r = idx / scw, j = idx % scw, n = n0 + r;
        if (n >= N) continue;
        float acc = 0.0f;
        for (int m = 0; m < CM; ++m) acc += sAs[r * CM + m] * sW2s[m * scw + j];
        float outs = acc * attr[n] * INV_SQRT96;
        float hs = sCos[r] * sc_s[(size_t)n * CM + j] + sSin[r] * outs;
        if (!final_layer) {
            if (j < CS) {
                out_a[(size_t)n * CS + j] = siluf_(hs);            // g_s
            } else {
                int u = j - CS;
                float gate = sigmoidf_(hs);
                out_b[(size_t)n * CM + u * 3 + 0] = gate * sHv[r * CM + u * 3 + 0];
                out_b[(size_t)n * CM + u * 3 + 1] = gate * sHv[r * CM + u * 3 + 1];
                out_b[(size_t)n * CM + u * 3 + 2] = gate * sHv[r * CM + u * 3 + 2];
            }
        } else {
            out_a[(size_t)n * 160 + j] = hs;                       // o_s
        }
    }
    if (final_layer) {
        for (int idx = tid; idx < 8 * CM; idx += 256) {
            int r = idx / CM, j = idx % CM, n = n0 + r;
            if (n >= N) continue;
            out_a[(size_t)n * 160 + CS + j] = sHv[r * CM + j];     // o_v
        }
    }
}

// ---------------------------------------------------------------------------
extern "C" void kernel_launch(void* const* d_in, const int* in_sizes, int n_in,
                              void* d_out, int out_size, void* d_ws, size_t ws_size,
                              hipStream_t stream)
{
    const float* node_features = (const float*)d_in[0];
    const float* node_attr     = (const float*)d_in[1];
    const int*   edge_src      = (const int*)  d_in[2];
    const int*   edge_dst      = (const int*)  d_in[3];
    const float* edge_attr     = (const float*)d_in[4];
    const float* edge_scalars  = (const float*)d_in[5];
    const float* p1_sc_s = (const float*)d_in[6];
    const float* p1_sc_v = (const float*)d_in[7];
    const float* p1_l1_s = (const float*)d_in[8];
    const float* p1_l1_v = (const float*)d_in[9];
    const float* p1_fa   = (const float*)d_in[10];
    const float* p1_fb   = (const float*)d_in[11];
    const float* p1_l2_s = (const float*)d_in[12];
    const float* p1_l2_v = (const float*)d_in[13];
    const float* p1_l3   = (const float*)d_in[14];
    const float* p2_sc_s = (const float*)d_in[15];
    const float* p2_sc_v = (const float*)d_in[16];
    const float* p2_l1_s = (const float*)d_in[17];
    const float* p2_l1_v = (const float*)d_in[18];
    const float* p2_fa   = (const float*)d_in[19];
    const float* p2_fb   = (const float*)d_in[20];
    const float* p2_l2_s = (const float*)d_in[21];
    const float* p2_l2_v = (const float*)d_in[22];
    const float* p2_l3   = (const float*)d_in[23];

    const int N = in_sizes[1];   // node_attr: N x 1
    const int E = in_sizes[2];   // edge_src: E

    // workspace layout (floats): ~898*N total (~72 MB)
    float* ws    = (float*)d_ws;
    float* deg   = ws;
    float* isd   = deg   + N;
    float* f_s   = isd   + N;
    float* f_v   = f_s   + (size_t)N * CS;
    float* sc_s  = f_v   + (size_t)N * CM;
    float* sc_v  = sc_s  + (size_t)N * CM;
    float* agg_s = sc_v  + (size_t)N * CM;
    float* agg_v = agg_s + (size_t)N * CM;          // N x 288, contiguous after agg_s
    float* g_s   = agg_v + (size_t)N * CM * 3;
    float* g_v   = g_s   + (size_t)N * CS;

    const int nbE  = (E + 255) / 256;
    const int nbN  = (N + 255) / 256;
    const int nbF  = (N + 15) / 16;
    const int nbB  = (N + 7) / 8;
    const int nbEd = (E + 63) / 64;

    // degrees (shared by both layers)
    hipMemsetAsync(deg, 0, (size_t)N * sizeof(float), stream);
    deg_kernel<<<nbE, 256, 0, stream>>>(edge_dst, deg, E);
    isd_kernel<<<nbN, 256, 0, stream>>>(deg, isd, N);

    // ---------------- layer 1 (scw = 96) ----------------
    node_front_kernel<<<nbF, 256, 0, stream>>>(
        node_features, CS + 3 * CV,            // s rows, stride 160
        node_features + CS, CS + 3 * CV,       // v rows, stride 160
        node_attr, p1_sc_s, CM, p1_sc_v, p1_l1_s, p1_l1_v,
        sc_s, sc_v, f_s, f_v, N);
    hipMemsetAsync(agg_s, 0, (size_t)N * (CM * 4) * sizeof(float), stream); // agg_s+agg_v
    edge_kernel<<<nbEd, 256, 0, stream>>>(
        edge_scalars, edge_attr, edge_src, edge_dst,
        f_s, f_v, p1_fa, p1_fb, agg_s, agg_v, E);
    node_back_kernel<<<nbB, 256, 0, stream>>>(
        agg_s, agg_v, isd, node_attr, sc_s, sc_v,
        p1_l2_s, CM, p1_l2_v, p1_l3,
        g_s, g_v, N, /*final=*/0);

    // ---------------- layer 2 (scw = 64) ----------------
    node_front_kernel<<<nbF, 256, 0, stream>>>(
        g_s, CS, g_v, CM,
        node_attr, p2_sc_s, CS, p2_sc_v, p2_l1_s, p2_l1_v,
        sc_s, sc_v, f_s, f_v, N);
    hipMemsetAsync(agg_s, 0, (size_t)N * (CM * 4) * sizeof(float), stream);
    edge_kernel<<<nbEd, 256, 0, stream>>>(
        edge_scalars, edge_attr, edge_src, edge_dst,
        f_s, f_v, p2_fa, p2_fb, agg_s, agg_v, E);
    node_back_kernel<<<nbB, 256, 0, stream>>>(
        agg_s, agg_v, isd, node_attr, sc_s, sc_v,
        p2_l2_s, CS, p2_l2_v, p2_l3,
        (float*)d_out, nullptr, N, /*final=*/1);
}